// MultiHeadAttentionLayer_82497731822089
// MI455X (gfx1250) — compile-verified
//
#include <hip/hip_runtime.h>

typedef float v2f __attribute__((ext_vector_type(2)));
typedef float v8f __attribute__((ext_vector_type(8)));

#define N_NODES 50000
#define N_EDGES 1600000
#define HD 64        // HEADS * DPH
#define WPITCH 68    // padded LDS pitch: conflict-free b64 B-fragment reads

// ---------------------------------------------------------------------------
// zero fill
// ---------------------------------------------------------------------------
__global__ __launch_bounds__(256) void zero_kernel(float* p, int n) {
    int i = blockIdx.x * 256 + threadIdx.x;
    if (i < n) p[i] = 0.0f;
}

// Cooperative load of W (64x64, row-major in memory) into LDS transposed:
// sWt[n*WPITCH + k] = W[k*64 + n].  B-fragment (K pair, fixed n) becomes one
// contiguous, 8B-aligned ds_load_b64.
__device__ __forceinline__ void load_weights_transposed(
    const float* __restrict__ W, const float* __restrict__ b,
    float* sWt, float* sb, int t) {
    for (int i = t; i < 64 * 64; i += 256) {
        const int k = i >> 6;
        const int n = i & 63;
        sWt[n * WPITCH + k] = W[i];
    }
    if (t < 64) sb[t] = b[t];
    __syncthreads();
}

// ---------------------------------------------------------------------------
// Phase 1: Q/K/V = x @ W{q,k,v} + b.  blockIdx.y selects the matrix.
// One wave computes a 16-row x 64-col tile using V_WMMA_F32_16X16X4_F32.
// ---------------------------------------------------------------------------
__global__ __launch_bounds__(256)
void qkv_gemm_kernel(const float* __restrict__ x,
                     const float* __restrict__ Wq, const float* __restrict__ bq,
                     const float* __restrict__ Wk, const float* __restrict__ bk,
                     const float* __restrict__ Wv, const float* __restrict__ bv,
                     float* __restrict__ qkv_base) {
    __shared__ float sWt[64 * WPITCH];
    __shared__ float sb[64];

    const int m = blockIdx.y;  // 0=Q 1=K 2=V
    const float* W = (m == 0) ? Wq : (m == 1) ? Wk : Wv;
    const float* b = (m == 0) ? bq : (m == 1) ? bk : bv;

    const int t = threadIdx.x;
    load_weights_transposed(W, b, sWt, sb, t);

    const int wave  = t >> 5;
    const int lane  = t & 31;
    const int tile  = blockIdx.x * 8 + wave;     // 16-node tile
    if (tile * 16 >= N_NODES) return;            // wave-uniform: EXEC stays full

    const int lm    = lane & 15;                 // M (A) / N (B,C) index
    const int khalf = lane >> 4;                 // K split for A/B fragments

    v8f acc[4] = {v8f{}, v8f{}, v8f{}, v8f{}};   // 4 N-tiles of 16

    const float* arow = x + (size_t)(tile * 16 + lm) * HD + 2 * khalf;
    // per-nt base into transposed weights: row n = nt*16+lm, col k + 2*khalf
    const float* wt0 = sWt + (0 * 16 + lm) * WPITCH + 2 * khalf;
    const float* wt1 = sWt + (1 * 16 + lm) * WPITCH + 2 * khalf;
    const float* wt2 = sWt + (2 * 16 + lm) * WPITCH + 2 * khalf;
    const float* wt3 = sWt + (3 * 16 + lm) * WPITCH + 2 * khalf;

    for (int k = 0; k < 64; k += 4) {
        v2f a = *(const v2f*)(arow + k);         // A[m][k+2*khalf .. +1]
        acc[0] = __builtin_amdgcn_wmma_f32_16x16x4_f32(
            false, a, false, *(const v2f*)(wt0 + k), (short)0, acc[0], false, false);
        acc[1] = __builtin_amdgcn_wmma_f32_16x16x4_f32(
            false, a, false, *(const v2f*)(wt1 + k), (short)0, acc[1], false, false);
        acc[2] = __builtin_amdgcn_wmma_f32_16x16x4_f32(
            false, a, false, *(const v2f*)(wt2 + k), (short)0, acc[2], false, false);
        acc[3] = __builtin_amdgcn_wmma_f32_16x16x4_f32(
            false, a, false, *(const v2f*)(wt3 + k), (short)0, acc[3], false, false);
    }

    float* outp = qkv_base + (size_t)m * ((size_t)N_NODES * HD);
#pragma unroll
    for (int nt = 0; nt < 4; ++nt) {
        const int n = nt * 16 + lm;
        const float bias = sb[n];
#pragma unroll
        for (int r = 0; r < 8; ++r) {
            const int row = tile * 16 + r + 8 * khalf;
            outp[(size_t)row * HD + n] = acc[nt][r] + bias;
        }
    }
}

// ---------------------------------------------------------------------------
// Phase 2 (fused): per 16-edge tile, WMMA-compute proj_e = edge_attr @ We + be
// in registers, then contract with gathered Q[dst]*K[src] (L2-resident) to get
// score[e,h]; write e_out, w = exp(clip(score)), atomic-accumulate z[dst,h].
// C-fragment lane (l, vgpr r) holds proj[edge = r + 8*(l>>4)][n = 16*nt + (l&15)]
// -> head = n>>3, d = n&7; 8-lane shfl_xor tree reduces over d.
// ---------------------------------------------------------------------------
__global__ __launch_bounds__(256)
void edge_score_kernel(const float* __restrict__ edge_attr,
                       const int*   __restrict__ edge_index,
                       const float* __restrict__ We, const float* __restrict__ be,
                       const float* __restrict__ Q,  const float* __restrict__ K,
                       float* __restrict__ e_out, float* __restrict__ w_ws,
                       float* __restrict__ z_ws) {
    __shared__ float sWt[64 * WPITCH];
    __shared__ float sb[64];

    const int t = threadIdx.x;
    load_weights_transposed(We, be, sWt, sb, t);

    const int wave  = t >> 5;
    const int lane  = t & 31;
    const int tile  = blockIdx.x * 8 + wave;     // 16-edge tile (exact fit)
    const int lm    = lane & 15;
    const int khalf = lane >> 4;

    v8f acc[4] = {v8f{}, v8f{}, v8f{}, v8f{}};

    const float* arow = edge_attr + (size_t)(tile * 16 + lm) * HD + 2 * khalf;
    const float* wt0 = sWt + (0 * 16 + lm) * WPITCH + 2 * khalf;
    const float* wt1 = sWt + (1 * 16 + lm) * WPITCH + 2 * khalf;
    const float* wt2 = sWt + (2 * 16 + lm) * WPITCH + 2 * khalf;
    const float* wt3 = sWt + (3 * 16 + lm) * WPITCH + 2 * khalf;

    for (int k = 0; k < 64; k += 4) {
        v2f a = *(const v2f*)(arow + k);
        acc[0] = __builtin_amdgcn_wmma_f32_16x16x4_f32(
            false, a, false, *(const v2f*)(wt0 + k), (short)0, acc[0], false, false);
        acc[1] = __builtin_amdgcn_wmma_f32_16x16x4_f32(
            false, a, false, *(const v2f*)(wt1 + k), (short)0, acc[1], false, false);
        acc[2] = __builtin_amdgcn_wmma_f32_16x16x4_f32(
            false, a, false, *(const v2f*)(wt2 + k), (short)0, acc[2], false, false);
        acc[3] = __builtin_amdgcn_wmma_f32_16x16x4_f32(
            false, a, false, *(const v2f*)(wt3 + k), (short)0, acc[3], false, false);
    }

    // src/dst for the tile's 16 edges, held in lanes 0..15
    const int e16  = tile * 16 + lm;
    const int srcl = edge_index[e16];
    const int dstl = edge_index[N_EDGES + e16];
    const float rs = 0.3535533905932738f;        // 1/sqrt(DPH)

#pragma unroll
    for (int nt = 0; nt < 4; ++nt) {
        const int n    = nt * 16 + lm;           // n -> head = n>>3, d = n&7
        const float bias = sb[n];
#pragma unroll
        for (int r = 0; r < 8; ++r) {
            const int er = r + 8 * khalf;        // edge row in tile (uniform per 8-lane group)
            const int s  = __shfl(srcl, er, 32);
            const int d  = __shfl(dstl, er, 32);
            const float p  = acc[nt][r] + bias;  // proj_e[er][n]
            const float qv = Q[(size_t)d * HD + n];
            const float kv = K[(size_t)s * HD + n];
            float part = p * qv * kv * rs;
            // reduce over d = n&7 (8 contiguous lanes share edge & head)
            part += __shfl_xor(part, 1, 32);
            part += __shfl_xor(part, 2, 32);
            part += __shfl_xor(part, 4, 32);
            if ((lane & 7) == 0) {
                const int head = 2 * nt + ((lane >> 3) & 1);
                const size_t eh = (size_t)(tile * 16 + er) * 8 + head;
                e_out[eh] = part;                              // raw score
                const float wv = __expf(fminf(fmaxf(part, -5.0f), 5.0f));
                w_ws[eh] = wv;
                atomicAdd(&z_ws[(size_t)d * 8 + head], wv);    // L2-resident
            }
        }
    }
}

// ---------------------------------------------------------------------------
// Phase 3: alpha = w / (z[dst]+1e-6); out[dst] += V[src] * alpha.
// One thread per (edge, head); 8 f32 atomics into L2-resident out.
// ---------------------------------------------------------------------------
__global__ __launch_bounds__(256)
void aggregate_kernel(const int* __restrict__ edge_index,
                      const float* __restrict__ V,
                      const float* __restrict__ w_ws,
                      const float* __restrict__ z_ws,
                      float* __restrict__ out) {
    const int tid = blockIdx.x * 256 + threadIdx.x;   // 12.8M = E*H exactly
    const int e = tid >> 3;
    const int h = tid & 7;
    const int s = edge_index[e];
    const int d = edge_index[N_EDGES + e];
    const float alpha = w_ws[tid] / (z_ws[(size_t)d * 8 + h] + 1e-6f);
    const float* vp = V + (size_t)s * HD + h * 8;
    float* op = out + (size_t)d * HD + h * 8;
#pragma unroll
    for (int i = 0; i < 8; ++i) atomicAdd(&op[i], vp[i] * alpha);
}

// ---------------------------------------------------------------------------
extern "C" void kernel_launch(void* const* d_in, const int* in_sizes, int n_in,
                              void* d_out, int out_size, void* d_ws, size_t ws_size,
                              hipStream_t stream) {
    const float* x          = (const float*)d_in[0];
    const int*   edge_index = (const int*)  d_in[1];
    const float* edge_attr  = (const float*)d_in[2];
    const float* Wq = (const float*)d_in[3];
    const float* bq = (const float*)d_in[4];
    const float* Wk = (const float*)d_in[5];
    const float* bk = (const float*)d_in[6];
    const float* Wv = (const float*)d_in[7];
    const float* bv = (const float*)d_in[8];
    const float* We = (const float*)d_in[9];
    const float* be = (const float*)d_in[10];

    float* out   = (float*)d_out;                       // [N, 64]
    float* e_out = out + (size_t)N_NODES * HD;          // [E, 8]

    // workspace layout (floats): Q | K | V | z | w  ~= 91 MB
    float* ws = (float*)d_ws;
    float* Q  = ws;
    float* Kp = ws + (size_t)N_NODES * HD;
    float* Vp = ws + 2 * (size_t)N_NODES * HD;
    float* z  = ws + 3 * (size_t)N_NODES * HD;
    float* w  = z + (size_t)N_NODES * 8;

    // zero accumulators (d_out is poisoned; atomics need clean state each call)
    zero_kernel<<<(N_NODES * HD + 255) / 256, 256, 0, stream>>>(out, N_NODES * HD);
    zero_kernel<<<(N_NODES * 8 + 255) / 256, 256, 0, stream>>>(z, N_NODES * 8);

    // Phase 1: QKV GEMMs (3125 row-tiles, 8 waves/block)
    qkv_gemm_kernel<<<dim3((3125 + 7) / 8, 3), 256, 0, stream>>>(
        x, Wq, bq, Wk, bk, Wv, bv, Q);

    // Phase 2: fused edge projection + attention score (100000 tiles / 8 waves)
    edge_score_kernel<<<N_EDGES / 128, 256, 0, stream>>>(
        edge_attr, edge_index, We, be, Q, Kp, e_out, w, z);

    // Phase 3: softmax-normalized message aggregation
    aggregate_kernel<<<(N_EDGES * 8) / 256, 256, 0, stream>>>(
        edge_index, Vp, w, z, out);
}